// DTWLoss1_32701880992227
// MI455X (gfx1250) — compile-verified
//
#include <hip/hip_runtime.h>
#include <math.h>

// DTW wavefront, single persistent workgroup on one WGP (gfx1250).
// Fully branch-free inner bodies: inactive slots compute/store INF, which
// doubles as the diagonal top sentinel. Phase-split k-loop with register-
// hoisted sequence values and register-carried own-cell dependency.
#define DTW_NMAX 4096
#define NTHREADS 1024
#define NSLOTS   4      // DTW_NMAX / NTHREADS

__global__ __launch_bounds__(NTHREADS, 1)
void dtw_wavefront_kernel(const float* __restrict__ s,
                          const float* __restrict__ t,
                          float* __restrict__ out,
                          int n) {
    __shared__ __align__(16) float sS[DTW_NMAX];
    __shared__ __align__(16) float sT[DTW_NMAX];
    // dbuf[b][r+1] holds the diagonal value at row r; [0] is the row -1
    // sentinel (INF); row n (index n+1) is a write-only scratch row for
    // inactive shrink-phase slots.
    __shared__ __align__(16) float dbuf[3][DTW_NMAX + 2];

    const int tid = threadIdx.x;
    const float INF = __builtin_inff();

    // ---- Stage both sequences into LDS via CDNA5 async global->LDS copy ----
    // 1024 threads x 16B per array; bypasses VGPRs, tracked with ASYNCcnt.
    {
        unsigned voff = (unsigned)tid * 16u;
        unsigned ldsS = (unsigned)(size_t)(&sS[0]) + voff;
        unsigned ldsT = (unsigned)(size_t)(&sT[0]) + voff;
        if ((tid * 4) < n) {
            asm volatile("global_load_async_to_lds_b128 %0, %1, %2"
                         :: "v"(ldsS), "v"(voff), "s"(s) : "memory");
            asm volatile("global_load_async_to_lds_b128 %0, %1, %2"
                         :: "v"(ldsT), "v"(voff), "s"(t) : "memory");
        }
        asm volatile("s_wait_asynccnt 0x0" ::: "memory");
    }
    __syncthreads();

    // Per-slot register state.
    float own[NSLOTS];   // this slot's DP value from the previous diagonal
    float regS[NSLOTS];  // sS[i], i constant per slot during growth phase
#pragma unroll
    for (int m = 0; m < NSLOTS; ++m) {
        int i = tid + m * NTHREADS;
        own[m]  = INF;                      // == DTW[i][-1] before activation
        regS[m] = (i < n) ? sS[i] : 0.0f;
    }

    // Seed diagonal 0 and sentinels (thread 0).
    if (tid == 0) {
        float d0  = sS[0] - sT[0];
        float c00 = d0 * d0;                // DTW[0][0]
        dbuf[0][0] = INF; dbuf[1][0] = INF; dbuf[2][0] = INF;  // row -1
        dbuf[0][1] = INF;                   // virtual diag -1, row 0
        dbuf[1][1] = c00;                   // diag 0, row 0
        dbuf[1][2] = INF;                   // diag 0 top sentinel (row 1)
        own[0] = c00;
    }
    __syncthreads();

    float* prev2 = dbuf[0];  // diagonal k-2
    float* prev  = dbuf[1];  // diagonal k-1
    float* cur   = dbuf[2];  // diagonal k

    // ---- Growth phase: k = 1 .. n-1, rows 0..k; row i fixed per slot ----
    // Inactive slots (i > k) write INF at index i+1; the slot at i = k+1
    // thereby writes the top sentinel (row k+1) that diag k+2 reads as dg.
    for (int k = 1; k < n; ++k) {
#pragma unroll
        for (int m = 0; m < NSLOTS; ++m) {
            int i = tid + m * NTHREADS;
            int j = k - i;
            int jc = (j > 0) ? j : 0;            // clamp; value unused if j<0
            float d  = regS[m] - sT[jc];
            float up = prev[i];     // DTW[i-1][j]   (row i-1 -> index i)
            float dg = prev2[i];    // DTW[i-1][j-1] (row i-1 on diag k-2)
            float v  = fminf(fminf(up, own[m]), dg);   // own = DTW[i][j-1]
            v = fmaf(d, d, v);
            v = (j >= 0) ? v : INF;              // branchless predication
            cur[i + 1] = v;
            own[m] = v;
        }
        __syncthreads();
        float* tmp = prev2; prev2 = prev; prev = cur; cur = tmp;
    }

    // Hoist sT for the shrink phase: j = n-1-slot is constant per slot.
    float regT[NSLOTS];
#pragma unroll
    for (int m = 0; m < NSLOTS; ++m) {
        int slot = tid + m * NTHREADS;
        regT[m] = (slot < n) ? sT[(n - 1) - slot] : 0.0f;
    }

    // ---- Shrink phase: k = n .. 2n-2, rows ilo..n-1; all cells interior ----
    // Inactive slots (i > n-1) clamp reads to row n-1 (discarded) and store
    // INF into scratch row n (index n+1), which no read touches.
    for (int k = n; k <= 2 * n - 2; ++k) {
        int ilo = k - n + 1;
#pragma unroll
        for (int m = 0; m < NSLOTS; ++m) {
            int slot = tid + m * NTHREADS;
            int i = ilo + slot;
            int ic = (i < n) ? i : (n - 1);      // clamped read index
            float d    = sS[ic] - regT[m];
            float left = prev[ic + 1];  // DTW[i][j-1]   (row i on diag k-1)
            float dg   = prev2[ic];     // DTW[i-1][j-1] (row i-1 on diag k-2)
            float v    = fminf(fminf(own[m], left), dg);  // own = DTW[i-1][j]
            v = fmaf(d, d, v);
            v = (i < n) ? v : INF;               // branchless predication
            int st = (i < n) ? (i + 1) : (n + 1);
            cur[st] = v;
            own[m] = v;
        }
        __syncthreads();
        float* tmp = prev2; prev2 = prev; prev = cur; cur = tmp;
    }

    // Last diagonal (single cell, row n-1) ended up in prev after rotation.
    if (tid == 0) {
        out[0] = __builtin_sqrtf(prev[n]);
    }
}

extern "C" void kernel_launch(void* const* d_in, const int* in_sizes, int n_in,
                              void* d_out, int out_size, void* d_ws, size_t ws_size,
                              hipStream_t stream) {
    (void)n_in; (void)d_ws; (void)ws_size; (void)out_size;
    const float* s = (const float*)d_in[0];
    const float* t = (const float*)d_in[1];
    float* out = (float*)d_out;
    int n = in_sizes[0];
    if (n > DTW_NMAX) n = DTW_NMAX;

    dtw_wavefront_kernel<<<dim3(1), dim3(NTHREADS), 0, stream>>>(s, t, out, n);
}